// KANC_CIFAR_23974507446557
// MI455X (gfx1250) — compile-verified
//
#include <hip/hip_runtime.h>
#include <math.h>

typedef __attribute__((ext_vector_type(2))) float v2f;
typedef __attribute__((ext_vector_type(8))) float v8f;

// ---------------------------------------------------------------------------
// KAN activation pieces: SiLU + the 6 cubic B-spline bases on the
// efficient-kan grid (GRID_SIZE=3, SPLINE_ORDER=3, knots t_j=(j-3)*(2/3)-1).
// ---------------------------------------------------------------------------
__device__ __forceinline__ float silu_f(float x) {
    return x / (1.0f + __expf(-x));
}

__device__ __forceinline__ void bspline6(float x, float* out6) {
    const float h = 2.0f / 3.0f;
    float bb[9];
#pragma unroll
    for (int j = 0; j < 9; ++j) {
        float t0 = (float)(j - 3) * h - 1.0f;
        float t1 = (float)(j - 2) * h - 1.0f;
        bb[j] = (x >= t0 && x < t1) ? 1.0f : 0.0f;
    }
#pragma unroll
    for (int p = 1; p <= 3; ++p) {
        float inv = 1.0f / ((float)p * h);
#pragma unroll
        for (int j = 0; j < 8; ++j) {
            if (j < 9 - p) {
                float tj   = (float)(j - 3) * h - 1.0f;
                float tjp1 = (float)(j + p - 2) * h - 1.0f;   // t_{j+p+1}
                bb[j] = ((x - tj) * bb[j] + (tjp1 - x) * bb[j + 1]) * inv;
            }
        }
    }
#pragma unroll
    for (int k = 0; k < 6; ++k) out6[k] = bb[k];
}

// ---------------------------------------------------------------------------
// Fold base_w / spline_w / scaler into one padded weight matrix
// Wc[n][f], n in [0, NT*16), f in [0, FP).  f = i*7 + s, s==0 -> base path.
// ---------------------------------------------------------------------------
template<int IN, int COUT, int NT, int FP>
__global__ void build_wc(const float* __restrict__ bw, const float* __restrict__ sw,
                         const float* __restrict__ sc, float* __restrict__ Wc) {
    int idx = blockIdx.x * blockDim.x + threadIdx.x;
    constexpr int TOTAL = NT * 16 * FP;
    if (idx >= TOTAL) return;
    int n = idx / FP;
    int f = idx - n * FP;
    constexpr int F = IN * 7;
    float v = 0.0f;
    if (n < COUT && f < F) {
        int i = f / 7;
        int s = f - i * 7;
        if (s == 0) v = bw[n * IN + i];
        else        v = sw[(n * IN + i) * 6 + (s - 1)] * sc[n * IN + i];
    }
    Wc[idx] = v;
}

// ---------------------------------------------------------------------------
// KAN conv3x3 (pad 1) as 16-patch x COUT tiles via V_WMMA_F32_16X16X4_F32.
// One wave (32 threads) per tile.  All NT=ceil(COUT/16) column tiles are
// accumulated in-wave so the expensive feature build (SiLU + B-splines into
// LDS) happens exactly once per patch tile.
// ---------------------------------------------------------------------------
template<int CIN, int COUT, int H, int W>
__global__ __launch_bounds__(32)
void kan_conv(const float* __restrict__ in, const float* __restrict__ Wc,
              float* __restrict__ out) {
    constexpr int IN = CIN * 9;
    constexpr int F  = IN * 7;
    constexpr int FP = (F + 3) & ~3;        // K padded to multiple of 4
    constexpr int FS = FP + 4;              // LDS row stride (bank spread)
    constexpr int NT = (COUT + 15) / 16;    // N tiles, accumulated in-wave

    __shared__ float feat[16][FS];

    const int lane  = threadIdx.x;
    const int tile  = blockIdx.x;
    const int pbase = tile * 16;

    // zero K padding (layer 1 only: FP-F == 3)
    if constexpr (FP > F) {
        constexpr int PAD = FP - F;
        for (int z = lane; z < 16 * PAD; z += 32) {
            int mm = z / PAD;
            int ff = F + (z - mm * PAD);
            feat[mm][ff] = 0.0f;
        }
    }

    // cooperative feature build: 16 patches x IN values -> 7 floats each
    for (int idx = lane; idx < 16 * IN; idx += 32) {
        int m = idx / IN;
        int i = idx - m * IN;
        int p = pbase + m;
        int b   = p / (H * W);
        int rem = p - b * (H * W);
        int y = rem / W;
        int x = rem - y * W;
        int c = i / 9;
        int r = i - c * 9;
        int dy = r / 3 - 1;
        int dx = (r - (r / 3) * 3) - 1;
        int yy = y + dy, xx = x + dx;
        float v = 0.0f;
        if (yy >= 0 && yy < H && xx >= 0 && xx < W)
            v = in[((b * CIN + c) * H + yy) * W + xx];
        float* dst = &feat[m][i * 7];
        dst[0] = silu_f(v);
        float bs[6];
        bspline6(v, bs);
#pragma unroll
        for (int k = 0; k < 6; ++k) dst[1 + k] = bs[k];
    }
    __syncthreads();

    // WMMA fragment coordinates (wave32, f32 16x16x4 layout)
    const int m  = lane & 15;   // A row / B column / D column
    const int kh = lane >> 4;   // K-half selector

    const float* brow[NT];
#pragma unroll
    for (int n = 0; n < NT; ++n)
        brow[n] = Wc + (n * 16 + m) * FP;

    v8f acc[NT];
#pragma unroll
    for (int n = 0; n < NT; ++n)
        acc[n] = (v8f){0.f, 0.f, 0.f, 0.f, 0.f, 0.f, 0.f, 0.f};

    for (int kk = 0; kk < FP; kk += 4) {
        v2f a = *(const v2f*)&feat[m][kk + 2 * kh];       // ds_load_b64
#pragma unroll
        for (int n = 0; n < NT; ++n) {
            v2f b = *(const v2f*)&brow[n][kk + 2 * kh];   // global_load_b64 (L2 hot)
            acc[n] = __builtin_amdgcn_wmma_f32_16x16x4_f32(
                false, a, false, b, (short)0, acc[n], false, false);
        }
    }

    // D layout: VGPR j -> M = j + 8*kh, N = lane&15
#pragma unroll
    for (int n = 0; n < NT; ++n) {
        int o = n * 16 + m;
        if (o < COUT) {
#pragma unroll
            for (int j = 0; j < 8; ++j) {
                int M = j + 8 * kh;
                int p = pbase + M;
                int b   = p / (H * W);
                int rem = p - b * (H * W);
                int y = rem / W;
                int x = rem - y * W;
                out[((b * COUT + o) * H + y) * W + x] = acc[n][j];
            }
        }
    }
}

// ---------------------------------------------------------------------------
// 2x2 max pool, NCHW
// ---------------------------------------------------------------------------
template<int C, int H, int W>
__global__ void maxpool2k(const float* __restrict__ in, float* __restrict__ out,
                          int total) {
    int idx = blockIdx.x * blockDim.x + threadIdx.x;
    if (idx >= total) return;
    constexpr int HO = H / 2, WO = W / 2;
    int x = idx % WO; int t = idx / WO;
    int y = t % HO;   t /= HO;
    int c = t % C;    int b = t / C;
    const float* base = in + ((b * C + c) * H + 2 * y) * W + 2 * x;
    float m0 = fmaxf(base[0], base[1]);
    float m1 = fmaxf(base[W], base[W + 1]);
    out[idx] = fmaxf(m0, m1);
}

// ---------------------------------------------------------------------------
// Final FC: (256,512) @ (10,512)^T + b  -> (256,10).  Tiny; plain VALU.
// ---------------------------------------------------------------------------
__global__ void fc_kernel(const float* __restrict__ in, const float* __restrict__ w,
                          const float* __restrict__ bias, float* __restrict__ out) {
    int idx = blockIdx.x * blockDim.x + threadIdx.x;
    if (idx >= 256 * 10) return;
    int o = idx % 10, b = idx / 10;
    const float* xi = in + b * 512;
    const float* wo = w + o * 512;
    float s = bias[o];
    for (int j = 0; j < 512; ++j) s += xi[j] * wo[j];
    out[idx] = s;
}

// ---------------------------------------------------------------------------
extern "C" void kernel_launch(void* const* d_in, const int* in_sizes, int n_in,
                              void* d_out, int out_size, void* d_ws, size_t ws_size,
                              hipStream_t stream) {
    (void)in_sizes; (void)n_in; (void)out_size; (void)ws_size;
    const float* x   = (const float*)d_in[0];
    const float* bw1 = (const float*)d_in[1];
    const float* sw1 = (const float*)d_in[2];
    const float* sc1 = (const float*)d_in[3];
    const float* bw2 = (const float*)d_in[4];
    const float* sw2 = (const float*)d_in[5];
    const float* sc2 = (const float*)d_in[6];
    const float* bw3 = (const float*)d_in[7];
    const float* sw3 = (const float*)d_in[8];
    const float* sc3 = (const float*)d_in[9];
    const float* fcw = (const float*)d_in[10];
    const float* fcb = (const float*)d_in[11];

    float* ws   = (float*)d_ws;
    float* Wc1  = ws;                       // 16*192  = 3072
    float* Wc2  = Wc1 + 16 * 192;           // 16*504  = 8064
    float* Wc3  = Wc2 + 16 * 504;           // 32*1008 = 32256
    float* conv = Wc3 + 32 * 1008;          // max 256*8*32*32 = 2097152
    float* plA  = conv + 256 * 8 * 32 * 32; // max 524288
    float* plB  = plA + 256 * 8 * 16 * 16;  // max 262144

    build_wc<27, 8, 1, 192  ><<<(16 * 192  + 255) / 256, 256, 0, stream>>>(bw1, sw1, sc1, Wc1);
    build_wc<72, 16, 1, 504 ><<<(16 * 504  + 255) / 256, 256, 0, stream>>>(bw2, sw2, sc2, Wc2);
    build_wc<144, 32, 2, 1008><<<(32 * 1008 + 255) / 256, 256, 0, stream>>>(bw3, sw3, sc3, Wc3);

    // layer 1: (256,3,32,32) -> conv (256,8,32,32) -> pool (256,8,16,16)
    kan_conv<3, 8, 32, 32><<<16384, 32, 0, stream>>>(x, Wc1, conv);
    maxpool2k<8, 32, 32><<<(256 * 8 * 16 * 16 + 255) / 256, 256, 0, stream>>>(conv, plA, 256 * 8 * 16 * 16);

    // layer 2: -> conv (256,16,16,16) -> pool (256,16,8,8)
    kan_conv<8, 16, 16, 16><<<4096, 32, 0, stream>>>(plA, Wc2, conv);
    maxpool2k<16, 16, 16><<<(256 * 16 * 8 * 8 + 255) / 256, 256, 0, stream>>>(conv, plB, 256 * 16 * 8 * 8);

    // layer 3: -> conv (256,32,8,8) (both N tiles in-wave) -> pool (256,32,4,4)
    kan_conv<16, 32, 8, 8><<<1024, 32, 0, stream>>>(plB, Wc3, conv);
    maxpool2k<32, 8, 8><<<(256 * 32 * 4 * 4 + 255) / 256, 256, 0, stream>>>(conv, plA, 256 * 32 * 4 * 4);

    // FC head
    fc_kernel<<<(2560 + 63) / 64, 64, 0, stream>>>(plA, fcw, fcb, (float*)d_out);
}